// PointNetSetAbstractionQ_69982197121137
// MI455X (gfx1250) — compile-verified
//
#include <hip/hip_runtime.h>
#include <math.h>

#define B_       16
#define N_       1024
#define DF_      64
#define NPOINT_  512
#define NSAMPLE_ 32
#define POS_     (NPOINT_*NSAMPLE_)   /* 16384 positions per b3 */
#define B3_      (B_*3)
#define RAD2_    0.04f
#define EPS_     1e-5f
#define CNT_     786432.0f            /* 48 * 16384 elements per channel */

typedef float v2f __attribute__((ext_vector_type(2)));
typedef float v8f __attribute__((ext_vector_type(8)));

// ---------------------------------------------------------------------------
// 1) Farthest point sampling: 1 block per batch, 1 thread per point.
// ---------------------------------------------------------------------------
__global__ __launch_bounds__(1024)
void fps_kernel(const float* __restrict__ xyz, int* __restrict__ fps_idx,
                float* __restrict__ nxyz_ws, float* __restrict__ out_xyz)
{
    __shared__ float rv[1024];
    __shared__ int   ri[1024];
    __shared__ float cpt[3];
    const int b = blockIdx.x, i = threadIdx.x;
    const float x = xyz[(b*3+0)*N_ + i];
    const float y = xyz[(b*3+1)*N_ + i];
    const float z = xyz[(b*3+2)*N_ + i];

    float cx, cy, cz;
    rv[i] = x; __syncthreads();
    for (int s = 512; s > 0; s >>= 1) { if (i < s) rv[i] += rv[i+s]; __syncthreads(); }
    cx = rv[0] * (1.0f/N_); __syncthreads();
    rv[i] = y; __syncthreads();
    for (int s = 512; s > 0; s >>= 1) { if (i < s) rv[i] += rv[i+s]; __syncthreads(); }
    cy = rv[0] * (1.0f/N_); __syncthreads();
    rv[i] = z; __syncthreads();
    for (int s = 512; s > 0; s >>= 1) { if (i < s) rv[i] += rv[i+s]; __syncthreads(); }
    cz = rv[0] * (1.0f/N_); __syncthreads();

    float dx = x-cx, dy = y-cy, dz = z-cz;
    float dist = fminf(1e10f, dx*dx + dy*dy + dz*dz);

    for (int t = 0; t < NPOINT_; ++t) {
        rv[i] = dist; ri[i] = i; __syncthreads();
        for (int s = 512; s > 0; s >>= 1) {
            if (i < s) {
                float ov = rv[i+s]; int oi = ri[i+s];
                if (ov > rv[i] || (ov == rv[i] && oi < ri[i])) { rv[i] = ov; ri[i] = oi; }
            }
            __syncthreads();
        }
        const int sel = ri[0];
        if (i == sel) {
            cpt[0] = x; cpt[1] = y; cpt[2] = z;
            fps_idx[b*NPOINT_ + t] = sel;
            nxyz_ws[(b*NPOINT_ + t)*3 + 0] = x;
            nxyz_ws[(b*NPOINT_ + t)*3 + 1] = y;
            nxyz_ws[(b*NPOINT_ + t)*3 + 2] = z;
            out_xyz[(b*3+0)*NPOINT_ + t] = x;
            out_xyz[(b*3+1)*NPOINT_ + t] = y;
            out_xyz[(b*3+2)*NPOINT_ + t] = z;
        }
        __syncthreads();
        float ex = x - cpt[0], ey = y - cpt[1], ez = z - cpt[2];
        dist = fminf(dist, ex*ex + ey*ey + ez*ez);
    }
}

// ---------------------------------------------------------------------------
// 2) Ball query: one wave32 per centroid; ballot + bit-scan keeps ascending
//    index order (== reference's sort-then-truncate). Pad with first index.
// ---------------------------------------------------------------------------
__global__ __launch_bounds__(32)
void ball_query_kernel(const float* __restrict__ xyz, const float* __restrict__ nxyz,
                       int* __restrict__ idx_buf)
{
    const int s = blockIdx.x, b = blockIdx.y, lane = threadIdx.x;
    const float cx = nxyz[(b*NPOINT_+s)*3 + 0];
    const float cy = nxyz[(b*NPOINT_+s)*3 + 1];
    const float cz = nxyz[(b*NPOINT_+s)*3 + 2];
    const int base_out = (b*NPOINT_ + s)*NSAMPLE_;

    int count = 0;
    for (int base = 0; base < N_ && count < NSAMPLE_; base += 32) {
        const int i = base + lane;
        const float dx = xyz[(b*3+0)*N_ + i] - cx;
        const float dy = xyz[(b*3+1)*N_ + i] - cy;
        const float dz = xyz[(b*3+2)*N_ + i] - cz;
        const float d2 = dx*dx + dy*dy + dz*dz;
        unsigned mask = __builtin_amdgcn_ballot_w32(d2 <= RAD2_);
        while (mask && count < NSAMPLE_) {
            const int j = __builtin_ctz(mask);
            if (lane == 0) idx_buf[base_out + count] = base + j;
            ++count;
            mask &= (mask - 1);
        }
    }
    if (lane == 0) {
        int first = (count > 0) ? idx_buf[base_out] : (N_ - 1);
        for (int k = count; k < NSAMPLE_; ++k) idx_buf[base_out + k] = first;
    }
}

// ---------------------------------------------------------------------------
// 3) Pad W0 (64x66) -> (64x68) with zero columns 66,67.
// ---------------------------------------------------------------------------
__global__ __launch_bounds__(256)
void pad_w0_kernel(const float* __restrict__ W0, float* __restrict__ Wp)
{
    for (int j = blockIdx.x*256 + threadIdx.x; j < 64*68; j += gridDim.x*256) {
        const int o = j / 68, c = j % 68;
        Wp[j] = (c < 66) ? W0[o*66 + c] : 0.0f;
    }
}

// ---------------------------------------------------------------------------
// 4) Build X0 in channel-pair interleaved layout:
//    element (b3, c, pos) at ((b3*34 + c/2)*POS_ + pos)*2 + (c&1).
//    pair 0 = {gn, gx}; pairs 1..32 = gathered point features; pair 33 = pad.
//    All stores are contiguous 8-byte (b64) per thread.
// ---------------------------------------------------------------------------
__global__ __launch_bounds__(256)
void build_x0_kernel(const float* __restrict__ xyz, const float* __restrict__ points,
                     const float* __restrict__ nxyz, const int* __restrict__ idx_buf,
                     float* __restrict__ X0)
{
    const int pos = blockIdx.x*256 + threadIdx.x;           // 0..16383
    const int b3 = blockIdx.y;
    const int b = b3 / 3, d = b3 % 3;
    const int s = pos >> 5, k = pos & 31;
    const int i = idx_buf[(b*NPOINT_ + s)*NSAMPLE_ + k];
    const float gx = xyz[(b*3 + d)*N_ + i];
    const float gn = gx - nxyz[(b*NPOINT_ + s)*3 + d];

    float* Xb = X0 + ((size_t)b3 * 34) * POS_ * 2 + (size_t)pos * 2;
    v2f p0; p0.x = gn; p0.y = gx;
    *(v2f*)Xb = p0;
    const float* prow = points + (size_t)b3 * DF_ * N_;
    #pragma unroll 8
    for (int c2 = 0; c2 < 32; ++c2) {
        v2f p; p.x = prow[(2*c2    )*N_ + i];
               p.y = prow[(2*c2 + 1)*N_ + i];
        *(v2f*)(Xb + (size_t)(c2 + 1) * POS_ * 2) = p;
    }
    v2f zz; zz.x = 0.0f; zz.y = 0.0f;
    *(v2f*)(Xb + (size_t)33 * POS_ * 2) = zz;                // pad channels 66,67
}

// ---------------------------------------------------------------------------
// 5) GEMM via fp32 WMMA (V_WMMA_F32_16X16X4_F32), fully unrolled, no branches.
//    X layout: channel-pair interleaved; W: padded row-major (stride CINPAD).
//    One wave: 16(o) x 64(pos) = 4 accumulator tiles sharing preloaded A.
//    A frag: lanes 0-15 M=lane K={4s,4s+1}; lanes 16-31 M=lane-16 K={4s+2,4s+3}.
//    B frag mirrored (one b64 load); C/D: VGPR r -> M = r + 8*(lane>=16), N=lane&15.
// ---------------------------------------------------------------------------
template<int CINPAD, int COUT>
__global__ __launch_bounds__(256)
void gemm_wmma_kernel(const float* __restrict__ X, const float* __restrict__ Wp,
                      float* __restrict__ Y)
{
    constexpr int KSTEPS = CINPAD / 4;
    constexpr int NT = 4;
    const int lane = threadIdx.x & 31;
    const int wave = threadIdx.x >> 5;
    const int o_base = wave * 16;
    const int pos_base = blockIdx.x * (16*NT);
    const int b3 = blockIdx.y;
    const int n  = lane & 15;
    const int kb = (lane >> 4) * 2;           // 0 or 2
    const int kp = kb >> 1;                   // 0 or 1 (pair offset)

    // Preload all A fragments (W is tiny, L0/L2 resident).
    v2f afrag[KSTEPS];
    const float* Wr = Wp + (size_t)(o_base + n) * CINPAD + kb;
    #pragma unroll
    for (int s = 0; s < KSTEPS; ++s)
        afrag[s] = *(const v2f*)(Wr + 4*s);

    const float* Xb = X + ((size_t)b3 * (CINPAD/2)) * POS_ * 2;

    v8f acc[NT];
    const v8f vz = {0.f,0.f,0.f,0.f,0.f,0.f,0.f,0.f};
    #pragma unroll
    for (int t = 0; t < NT; ++t) acc[t] = vz;

    #pragma unroll
    for (int t = 0; t < NT; ++t) {
        const int pos = pos_base + t*16 + n;
        #pragma unroll
        for (int s = 0; s < KSTEPS; ++s) {
            const v2f bf = *(const v2f*)(Xb + ((size_t)(2*s + kp) * POS_ + pos) * 2);
            acc[t] = __builtin_amdgcn_wmma_f32_16x16x4_f32(
                         false, afrag[s], false, bf, (short)0, acc[t], false, false);
        }
    }

    const int mh = (lane >> 4) * 8;
    #pragma unroll
    for (int t = 0; t < NT; ++t) {
        float* Yb = Y + ((size_t)b3 * COUT + o_base + mh) * POS_ + pos_base + t*16 + n;
        #pragma unroll
        for (int r = 0; r < 8; ++r)
            Yb[(size_t)r * POS_] = acc[t][r];
    }
}

// ---------------------------------------------------------------------------
// 6) Per-channel sum of squares over plane-layout Y (for _qbn global RMS).
// ---------------------------------------------------------------------------
__global__ __launch_bounds__(256)
void sumsq_kernel(const float* __restrict__ Y, float* __restrict__ sumsq, int Cout)
{
    __shared__ float sm[256];
    const int c = blockIdx.x, b3 = blockIdx.y;
    const float* p = Y + ((size_t)b3 * Cout + c) * POS_;
    float acc = 0.0f;
    for (int j = threadIdx.x; j < POS_; j += 256) { const float v = p[j]; acc += v*v; }
    sm[threadIdx.x] = acc; __syncthreads();
    for (int s = 128; s > 0; s >>= 1) {
        if (threadIdx.x < s) sm[threadIdx.x] += sm[threadIdx.x + s];
        __syncthreads();
    }
    if (threadIdx.x == 0) atomicAdd(&sumsq[c], sm[0]);
}

__global__ void zero_kernel(float* __restrict__ p, int n)
{
    const int i = blockIdx.x * blockDim.x + threadIdx.x;
    if (i < n) p[i] = 0.0f;
}

// ---------------------------------------------------------------------------
// 7) Fused _qbn + _qrelu, one thread per (b, channel-pair, pos).
//    Reads plane-layout Y, writes interleaved Xn with v2f (b64) stores.
// ---------------------------------------------------------------------------
__global__ __launch_bounds__(256)
void qbn_qrelu_kernel(const float* __restrict__ Y, float* __restrict__ Xn,
                      const float* __restrict__ sumsq, int C)
{
    const int pos = blockIdx.x*256 + threadIdx.x;
    const int c2 = blockIdx.y;                // channel pair
    const int b = blockIdx.z;
    const int c = 2*c2;
    const float invA = rsqrtf(sumsq[c    ] * (1.0f / CNT_) + EPS_);
    const float invB = rsqrtf(sumsq[c + 1] * (1.0f / CNT_) + EPS_);

    float za0, za1, za2, zb0, zb1, zb2;
    {
        const size_t r0 = ((size_t)(3*b+0)*C + c)*POS_ + pos;
        const size_t r1 = ((size_t)(3*b+1)*C + c)*POS_ + pos;
        const size_t r2 = ((size_t)(3*b+2)*C + c)*POS_ + pos;
        za0 = Y[r0]*invA; za1 = Y[r1]*invA; za2 = Y[r2]*invA;
        zb0 = Y[r0+POS_]*invB; zb1 = Y[r1+POS_]*invB; zb2 = Y[r2+POS_]*invB;
    }
    const float modA = sqrtf(za0*za0 + za1*za1 + za2*za2);
    const float modB = sqrtf(zb0*zb0 + zb1*zb1 + zb2*zb2);
    const float cA = modA / fmaxf(modA, 1.0f);
    const float cB = modB / fmaxf(modB, 1.0f);

    const size_t base = ((size_t)c2 * POS_ + pos) * 2;
    const size_t bstr = (size_t)(C/2) * POS_ * 2;
    v2f o0; o0.x = cA*za0; o0.y = cB*zb0;
    v2f o1; o1.x = cA*za1; o1.y = cB*zb1;
    v2f o2; o2.x = cA*za2; o2.y = cB*zb2;
    *(v2f*)(Xn + (size_t)(3*b+0)*bstr + base) = o0;
    *(v2f*)(Xn + (size_t)(3*b+1)*bstr + base) = o1;
    *(v2f*)(Xn + (size_t)(3*b+2)*bstr + base) = o2;
}

// ---------------------------------------------------------------------------
// 8) Quaternion maxpool over k; input is interleaved (64 pairs of 128 ch).
// ---------------------------------------------------------------------------
__global__ __launch_bounds__(256)
void qmaxpool_kernel(const float* __restrict__ A, float* __restrict__ out)
{
    const int s = blockIdx.x*256 + threadIdx.x;   // 0..511
    const int c = blockIdx.y;                     // 0..127
    const int b = blockIdx.z;                     // 0..15
    const size_t bstr = (size_t)64 * POS_ * 2;
    const size_t base = ((size_t)(c >> 1) * POS_ + (size_t)s * NSAMPLE_) * 2 + (c & 1);
    const float* p0 = A + (size_t)(3*b+0)*bstr + base;
    const float* p1 = A + (size_t)(3*b+1)*bstr + base;
    const float* p2 = A + (size_t)(3*b+2)*bstr + base;
    float best = -1.0f; int bk = 0;
    for (int k = 0; k < NSAMPLE_; ++k) {
        const float v0 = p0[2*k], v1 = p1[2*k], v2 = p2[2*k];
        const float m2 = v0*v0 + v1*v1 + v2*v2;
        if (m2 > best) { best = m2; bk = k; }
    }
    out[((size_t)(3*b+0)*128 + c)*NPOINT_ + s] = p0[2*bk];
    out[((size_t)(3*b+1)*128 + c)*NPOINT_ + s] = p1[2*bk];
    out[((size_t)(3*b+2)*128 + c)*NPOINT_ + s] = p2[2*bk];
}

// ---------------------------------------------------------------------------
extern "C" void kernel_launch(void* const* d_in, const int* in_sizes, int n_in,
                              void* d_out, int out_size, void* d_ws, size_t ws_size,
                              hipStream_t stream)
{
    (void)in_sizes; (void)n_in; (void)out_size; (void)ws_size;
    const float* xyz    = (const float*)d_in[0];   // (16, 3, 1024)
    const float* points = (const float*)d_in[1];   // (48, 64, 1024)
    const float* W0     = (const float*)d_in[2];   // (64, 66)
    const float* W1     = (const float*)d_in[3];   // (64, 64)
    const float* W2     = (const float*)d_in[4];   // (128, 64)
    float* out = (float*)d_out;                    // 24576 + 48*128*512

    char* ws = (char*)d_ws;
    size_t off = 0;
    float* sumsq   = (float*)(ws + off); off += 256 * sizeof(float);
    int*   fps_idx = (int*)  (ws + off); off += (size_t)B_*NPOINT_*sizeof(int);
    float* nxyz    = (float*)(ws + off); off += (size_t)B_*NPOINT_*3*sizeof(float);
    int*   idx_buf = (int*)  (ws + off); off += (size_t)B_*NPOINT_*NSAMPLE_*sizeof(int);
    float* w0pad   = (float*)(ws + off); off += (size_t)64*68*sizeof(float);
    off = (off + 255) & ~(size_t)255;
    const size_t actBytes = (size_t)B3_ * 128 * POS_ * sizeof(float);   // 402.7 MB
    float* bufA = (float*)(ws + off); off += actBytes;                  // X / activated
    float* bufB = (float*)(ws + off); off += actBytes;                  // GEMM output

    // Stage 1-3: sampling, grouping, input tensor, weight pad
    fps_kernel<<<B_, 1024, 0, stream>>>(xyz, fps_idx, nxyz, out);
    pad_w0_kernel<<<4, 256, 0, stream>>>(W0, w0pad);
    ball_query_kernel<<<dim3(NPOINT_, B_), 32, 0, stream>>>(xyz, nxyz, idx_buf);
    build_x0_kernel<<<dim3(POS_/256, B3_), 256, 0, stream>>>(xyz, points, nxyz, idx_buf, bufA);

    // Layer 0: Cin 66 (padded 68) -> 64
    zero_kernel<<<1, 128, 0, stream>>>(sumsq, 128);
    gemm_wmma_kernel<68, 64><<<dim3(POS_/64, B3_), 128, 0, stream>>>(bufA, w0pad, bufB);
    sumsq_kernel<<<dim3(64, B3_), 256, 0, stream>>>(bufB, sumsq, 64);
    qbn_qrelu_kernel<<<dim3(POS_/256, 32, B_), 256, 0, stream>>>(bufB, bufA, sumsq, 64);

    // Layer 1: 64 -> 64
    zero_kernel<<<1, 128, 0, stream>>>(sumsq, 128);
    gemm_wmma_kernel<64, 64><<<dim3(POS_/64, B3_), 128, 0, stream>>>(bufA, W1, bufB);
    sumsq_kernel<<<dim3(64, B3_), 256, 0, stream>>>(bufB, sumsq, 64);
    qbn_qrelu_kernel<<<dim3(POS_/256, 32, B_), 256, 0, stream>>>(bufB, bufA, sumsq, 64);

    // Layer 2: 64 -> 128
    zero_kernel<<<1, 128, 0, stream>>>(sumsq, 128);
    gemm_wmma_kernel<64, 128><<<dim3(POS_/64, B3_), 256, 0, stream>>>(bufA, W2, bufB);
    sumsq_kernel<<<dim3(128, B3_), 256, 0, stream>>>(bufB, sumsq, 128);
    qbn_qrelu_kernel<<<dim3(POS_/256, 64, B_), 256, 0, stream>>>(bufB, bufA, sumsq, 128);

    // Pool + emit features
    qmaxpool_kernel<<<dim3(2, 128, B_), 256, 0, stream>>>(bufA, out + (size_t)B_*3*NPOINT_);
}